// SimpleGNN_66400194396708
// MI455X (gfx1250) — compile-verified
//
#include <hip/hip_runtime.h>
#include <math.h>

typedef __attribute__((ext_vector_type(2))) float v2f;
typedef __attribute__((ext_vector_type(8))) float v8f;

#define DIM      256
#define MT       64    // rows per workgroup (amortizes W reads 4x vs 16)
#define LSTRIDE  258   // padded LDS stride for A operands (conflict-free b64 reads)

// ---------------------------------------------------------------- utilities
__global__ __launch_bounds__(256) void zero_i32(int* __restrict__ p, int n) {
    int i = blockIdx.x * 256 + threadIdx.x;
    int st = gridDim.x * 256;
    for (; i < n; i += st) p[i] = 0;
}

__global__ __launch_bounds__(256) void count_dst(const int* __restrict__ dst,
                                                 int* __restrict__ cnt, int nE) {
    int i = blockIdx.x * 256 + threadIdx.x;
    int st = gridDim.x * 256;
    for (; i < nE; i += st) atomicAdd(&cnt[dst[i]], 1);
}

// Hillis-Steele block scan: off[i] = exclusive prefix within block, bsum[b] = block total
__global__ __launch_bounds__(256) void scan_block(const int* __restrict__ cnt,
                                                  int* __restrict__ off,
                                                  int* __restrict__ bsum, int n) {
    __shared__ int s[256];
    const int tid = threadIdx.x;
    const int gid = blockIdx.x * 256 + tid;
    int v = (gid < n) ? cnt[gid] : 0;
    s[tid] = v;
    __syncthreads();
    for (int d = 1; d < 256; d <<= 1) {
        int t = (tid >= d) ? s[tid - d] : 0;
        __syncthreads();
        s[tid] += t;
        __syncthreads();
    }
    if (gid < n) off[gid] = s[tid] - v;
    if (tid == 255) bsum[blockIdx.x] = s[255];
}

// single-block scan of the block sums (in place -> exclusive)
__global__ __launch_bounds__(256) void scan_sums(int* __restrict__ bsum, int nb) {
    __shared__ int s[256];
    const int tid = threadIdx.x;
    int v = (tid < nb) ? bsum[tid] : 0;
    s[tid] = v;
    __syncthreads();
    for (int d = 1; d < 256; d <<= 1) {
        int t = (tid >= d) ? s[tid - d] : 0;
        __syncthreads();
        s[tid] += t;
        __syncthreads();
    }
    bsum[tid] = s[tid] - v;
}

// finalize offsets, seed cursors, write sentinel off[n] = nE
__global__ __launch_bounds__(256) void scan_finish(int* __restrict__ off,
                                                   const int* __restrict__ bsum,
                                                   int* __restrict__ cursor,
                                                   int n, int nE) {
    const int gid = blockIdx.x * 256 + threadIdx.x;
    if (gid < n) {
        int o = off[gid] + bsum[blockIdx.x];
        off[gid] = o;
        cursor[gid] = o;
    }
    if (gid == 0) off[n] = nE;
}

__global__ __launch_bounds__(256) void build_perm(const int* __restrict__ src,
                                                  const int* __restrict__ dst,
                                                  int* __restrict__ cursor,
                                                  int* __restrict__ perm, int nE) {
    int i = blockIdx.x * 256 + threadIdx.x;
    int st = gridDim.x * 256;
    for (; i < nE; i += st) {
        int slot = atomicAdd(&cursor[dst[i]], 1);
        perm[slot] = src[i];
    }
}

// -------------------------------------------------- neighbor-mean gather
// one wave32 per destination node; lane covers 8 of 256 columns (2 x float4)
__global__ __launch_bounds__(256) void gather_mean(const float* __restrict__ xsrc,
                                                   const int* __restrict__ perm,
                                                   const int* __restrict__ off,
                                                   float* __restrict__ meanOut,
                                                   int nDst) {
    const int wave = blockIdx.x * 8 + (threadIdx.x >> 5);
    const int lane = threadIdx.x & 31;
    if (wave >= nDst) return;
    const int s = off[wave], e = off[wave + 1];
    float a0 = 0.f, a1 = 0.f, a2 = 0.f, a3 = 0.f;
    float b0 = 0.f, b1 = 0.f, b2 = 0.f, b3 = 0.f;
    for (int j = s; j < e; ++j) {
        const float4* row = (const float4*)(xsrc + (size_t)perm[j] * DIM);
        float4 a = row[lane];
        float4 b = row[lane + 32];
        a0 += a.x; a1 += a.y; a2 += a.z; a3 += a.w;
        b0 += b.x; b1 += b.y; b2 += b.z; b3 += b.w;
    }
    const float inv = 1.0f / fmaxf((float)(e - s), 1.0f);
    float4* orow = (float4*)(meanOut + (size_t)wave * DIM);
    orow[lane]      = make_float4(a0 * inv, a1 * inv, a2 * inv, a3 * inv);
    orow[lane + 32] = make_float4(b0 * inv, b1 * inv, b2 * inv, b3 * inv);
}

// -------------------------------------------------- fused SAGE layer GEMM
// out = mean @ Wl^T + bl + x_self @ Wr^T ; optional row l2norm + relu.
// One WG = 64 output rows x 256 cols; result overwrites the mean buffer in place.
// Each wave owns 2 column tiles and 4 row strips -> each W fragment feeds 4 WMMAs.
__global__ __launch_bounds__(256) void sage_gemm(float* __restrict__ meanio,
                                                 const float* __restrict__ xdst,
                                                 const float* __restrict__ Wl,
                                                 const float* __restrict__ bl,
                                                 const float* __restrict__ Wr,
                                                 int relu_norm, int nRows) {
    __shared__ float sMean[MT * LSTRIDE];   // A operand; reused as output staging
    __shared__ float sSelf[MT * LSTRIDE];
    __shared__ float sPart[MT * 4];
    __shared__ float sInv [MT];

    const int tid = threadIdx.x;
    const int m0  = blockIdx.x * MT;

    // stage 64 rows of mean & self features into LDS (clamp tail rows; they stay
    // inside this WG's own strip, so the in-place update remains race-free)
    for (int i = tid; i < MT * DIM / 4; i += 256) {
        const int m = i >> 6;           // 64 float4 per row
        const int c = (i & 63) * 4;
        int gm = m0 + m;
        if (gm >= nRows) gm = nRows - 1;
        float4 a = *(const float4*)(meanio + (size_t)gm * DIM + c);
        float4 s = *(const float4*)(xdst   + (size_t)gm * DIM + c);
        float* pm = &sMean[m * LSTRIDE + c];
        pm[0] = a.x; pm[1] = a.y; pm[2] = a.z; pm[3] = a.w;
        float* ps = &sSelf[m * LSTRIDE + c];
        ps[0] = s.x; ps[1] = s.y; ps[2] = s.z; ps[3] = s.w;
    }
    __syncthreads();

    const int lane  = tid & 31;
    const int wv    = tid >> 5;
    const int mrow  = lane & 15;            // A: M index within a strip
    const int khalf = (lane >> 4) << 1;     // A/B: K base 0 or 2 per lane half
    const int ncol  = lane & 15;            // B/C/D: N index
    const int n0    = wv * 32;              // this wave's two 16-col tiles

    const float* wl0 = Wl + (size_t)(n0 + ncol) * DIM;       // B[k][n] = W[n][k]
    const float* wl1 = Wl + (size_t)(n0 + 16 + ncol) * DIM;
    const float* wr0 = Wr + (size_t)(n0 + ncol) * DIM;
    const float* wr1 = Wr + (size_t)(n0 + 16 + ncol) * DIM;

    v8f acc[4][2];
#pragma unroll
    for (int s = 0; s < 4; ++s)
#pragma unroll
        for (int t = 0; t < 2; ++t)
            acc[s][t] = (v8f){0.f, 0.f, 0.f, 0.f, 0.f, 0.f, 0.f, 0.f};

#pragma unroll 2
    for (int k = 0; k < DIM; k += 4) {
        const int kk = k + khalf;
        v2f l0 = *(const v2f*)(wl0 + kk);
        v2f l1 = *(const v2f*)(wl1 + kk);
        v2f r0 = *(const v2f*)(wr0 + kk);
        v2f r1 = *(const v2f*)(wr1 + kk);
#pragma unroll
        for (int s = 0; s < 4; ++s) {
            v2f aM = *(const v2f*)(&sMean[(s * 16 + mrow) * LSTRIDE] + kk);
            v2f aS = *(const v2f*)(&sSelf[(s * 16 + mrow) * LSTRIDE] + kk);
            acc[s][0] = __builtin_amdgcn_wmma_f32_16x16x4_f32(false, aM, false, l0, (short)0, acc[s][0], false, false);
            acc[s][0] = __builtin_amdgcn_wmma_f32_16x16x4_f32(false, aS, false, r0, (short)0, acc[s][0], false, false);
            acc[s][1] = __builtin_amdgcn_wmma_f32_16x16x4_f32(false, aM, false, l1, (short)0, acc[s][1], false, false);
            acc[s][1] = __builtin_amdgcn_wmma_f32_16x16x4_f32(false, aS, false, r1, (short)0, acc[s][1], false, false);
        }
    }

    // bias: every accumulator slot in a lane shares one output column
    const float bb0 = bl[n0 + ncol];
    const float bb1 = bl[n0 + 16 + ncol];
#pragma unroll
    for (int s = 0; s < 4; ++s)
#pragma unroll
        for (int r = 0; r < 8; ++r) { acc[s][0][r] += bb0; acc[s][1][r] += bb1; }

    __syncthreads();   // all waves finished reading A operands; sMean reusable

    // stage result tiles into sMean (C/D layout: vgpr r -> M=r / M=r+8 per lane half)
    const int mbase = (lane >> 4) << 3;
#pragma unroll
    for (int s = 0; s < 4; ++s)
#pragma unroll
        for (int r = 0; r < 8; ++r) {
            const int m = s * 16 + mbase + r;
            sMean[m * LSTRIDE + n0 + ncol]      = acc[s][0][r];
            sMean[m * LSTRIDE + n0 + 16 + ncol] = acc[s][1][r];
        }
    __syncthreads();

    if (relu_norm) {
        {                               // 4 partials per row: 64 rows x 4 segs = 256 threads
            const int m = tid >> 2, seg = tid & 3;
            const float* row = &sMean[m * LSTRIDE + seg * 64];
            float ss = 0.f;
#pragma unroll 16
            for (int c = 0; c < 64; ++c) ss += row[c] * row[c];
            sPart[m * 4 + seg] = ss;
        }
        __syncthreads();
        if (tid < MT) {
            const float ss = sPart[tid * 4] + sPart[tid * 4 + 1] +
                             sPart[tid * 4 + 2] + sPart[tid * 4 + 3];
            sInv[tid] = 1.0f / fmaxf(sqrtf(ss), 1e-12f);
        }
        __syncthreads();
    }

    // write back in place over the mean buffer: row m = tid>>2, 64-col segment
    {
        const int m  = tid >> 2;
        const int cb = (tid & 3) * 64;
        if (m0 + m < nRows) {
            const float inv = relu_norm ? sInv[m] : 1.0f;
            float* orow = meanio + (size_t)(m0 + m) * DIM;
#pragma unroll
            for (int c = 0; c < 64; c += 4) {
                float x0 = sMean[m * LSTRIDE + cb + c + 0];
                float x1 = sMean[m * LSTRIDE + cb + c + 1];
                float x2 = sMean[m * LSTRIDE + cb + c + 2];
                float x3 = sMean[m * LSTRIDE + cb + c + 3];
                if (relu_norm) {
                    x0 = fmaxf(x0 * inv, 0.f); x1 = fmaxf(x1 * inv, 0.f);
                    x2 = fmaxf(x2 * inv, 0.f); x3 = fmaxf(x3 * inv, 0.f);
                }
                *(float4*)(orow + cb + c) = make_float4(x0, x1, x2, x3);
            }
        }
    }
}

// ---------------------------------------------------------------- launcher
extern "C" void kernel_launch(void* const* d_in, const int* in_sizes, int n_in,
                              void* d_out, int out_size, void* d_ws, size_t ws_size,
                              hipStream_t stream) {
    (void)n_in; (void)out_size; (void)ws_size;
    const float* xA = (const float*)d_in[0];
    const float* xP = (const float*)d_in[1];
    const float* Wl = (const float*)d_in[2];
    const float* bl = (const float*)d_in[3];
    const float* Wr = (const float*)d_in[4];
    const int* eap  = (const int*)d_in[5];
    const int* epa  = (const int*)d_in[6];

    const int NA = in_sizes[0] / DIM;           // 50000
    const int NP = in_sizes[1] / DIM;           // 50000
    const int E  = in_sizes[5] / 2;             // 800000
    const int L  = in_sizes[2] / (2 * DIM * DIM);

    const size_t FA = (size_t)NA * DIM, FP = (size_t)NP * DIM;
    char* ws = (char*)d_ws;
    float* A0 = (float*)ws;  ws += FA * sizeof(float);
    float* A1 = (float*)ws;  ws += FA * sizeof(float);
    float* P0 = (float*)ws;  ws += FP * sizeof(float);
    float* P1 = (float*)ws;  ws += FP * sizeof(float);
    int* cntP  = (int*)ws;   ws += (size_t)NP * 4;
    int* offP  = (int*)ws;   ws += (size_t)(NP + 1) * 4;
    int* curP  = (int*)ws;   ws += (size_t)NP * 4;
    int* permP = (int*)ws;   ws += (size_t)E * 4;
    int* cntA  = (int*)ws;   ws += (size_t)NA * 4;
    int* offA  = (int*)ws;   ws += (size_t)(NA + 1) * 4;
    int* curA  = (int*)ws;   ws += (size_t)NA * 4;
    int* permA = (int*)ws;   ws += (size_t)E * 4;
    int* bsum  = (int*)ws;   ws += 256 * 4;

    const int nbP = (NP + 255) / 256, nbA = (NA + 255) / 256;

    // CSR for author->paper edges (paper destinations)
    zero_i32   <<<64, 256, 0, stream>>>(cntP, NP);
    count_dst  <<<1024, 256, 0, stream>>>(eap + E, cntP, E);
    scan_block <<<nbP, 256, 0, stream>>>(cntP, offP, bsum, NP);
    scan_sums  <<<1, 256, 0, stream>>>(bsum, nbP);
    scan_finish<<<nbP, 256, 0, stream>>>(offP, bsum, curP, NP, E);
    build_perm <<<1024, 256, 0, stream>>>(eap, eap + E, curP, permP, E);

    // CSR for paper->author edges (author destinations)
    zero_i32   <<<64, 256, 0, stream>>>(cntA, NA);
    count_dst  <<<1024, 256, 0, stream>>>(epa + E, cntA, E);
    scan_block <<<nbA, 256, 0, stream>>>(cntA, offA, bsum, NA);
    scan_sums  <<<1, 256, 0, stream>>>(bsum, nbA);
    scan_finish<<<nbA, 256, 0, stream>>>(offA, bsum, curA, NA, E);
    build_perm <<<1024, 256, 0, stream>>>(epa, epa + E, curA, permA, E);

    const float* xa_cur = xA;
    const float* xp_cur = xP;
    for (int l = 0; l < L; ++l) {
        const bool last = (l == L - 1);
        float* xa_next = last ? (float*)d_out        : ((l & 1) ? A1 : A0);
        float* xp_next = last ? (float*)d_out + FA   : ((l & 1) ? P1 : P0);

        gather_mean<<<(NP + 7) / 8, 256, 0, stream>>>(xa_cur, permP, offP, xp_next, NP);
        gather_mean<<<(NA + 7) / 8, 256, 0, stream>>>(xp_cur, permA, offA, xa_next, NA);

        const float* Wl0 = Wl + ((size_t)(l * 2 + 0)) * DIM * DIM;
        const float* Wl1 = Wl + ((size_t)(l * 2 + 1)) * DIM * DIM;
        const float* Wr0 = Wr + ((size_t)(l * 2 + 0)) * DIM * DIM;
        const float* Wr1 = Wr + ((size_t)(l * 2 + 1)) * DIM * DIM;
        const float* bl0 = bl + (size_t)(l * 2 + 0) * DIM;
        const float* bl1 = bl + (size_t)(l * 2 + 1) * DIM;

        sage_gemm<<<(NP + MT - 1) / MT, 256, 0, stream>>>(xp_next, xp_cur, Wl0, bl0, Wr0, last ? 0 : 1, NP);
        sage_gemm<<<(NA + MT - 1) / MT, 256, 0, stream>>>(xa_next, xa_cur, Wl1, bl1, Wr1, last ? 0 : 1, NA);

        xa_cur = xa_next;
        xp_cur = xp_next;
    }
}